// ConsciousnessGAN_37752762532597
// MI455X (gfx1250) — compile-verified
//
#include <hip/hip_runtime.h>
#include <hip/hip_bf16.h>

// ConsciousnessGAN forward for MI455X (gfx1250, wave32).
// Bandwidth-bound on gen_W3 (512MB) + disc_W1 (512MB, shared across d_real/d_fake).
// Small batch-32 GEMMs use V_WMMA_F32_16X16X4_F32 (fp32 in/out, no precision loss).

typedef float v2f __attribute__((ext_vector_type(2)));
typedef float v8f __attribute__((ext_vector_type(8)));

#define NCELLS 1024
#define HID    256
#define FLATD  (NCELLS * HID)   // 262144

__device__ __forceinline__ float warp_sum(float v) {
#pragma unroll
  for (int o = 16; o > 0; o >>= 1) v += __shfl_xor(v, o, 32);
  return v;
}

__device__ __forceinline__ v8f wmma4(v2f a, v2f b, v8f c) {
  // D(16x16,f32) = A(16x4,f32) x B(4x16,f32) + C
  return __builtin_amdgcn_wmma_f32_16x16x4_f32(
      false, a, false, b, (short)0, c, false, false);
}

// ---------------- K1: generator front MLP (tiny) -------------------------
__global__ void k1_gen_front(const float* __restrict__ x, const float* __restrict__ noise,
                             const float* __restrict__ W1, const float* __restrict__ b1,
                             const float* __restrict__ W2, const float* __restrict__ b2,
                             float* __restrict__ h2ws) {
  __shared__ float gi[288];
  __shared__ float h1[256];
  int t = threadIdx.x;  // 256 threads
  gi[t] = x[t];
  if (t < 32) gi[256 + t] = noise[t];
  __syncthreads();
  float s = b1[t];
  for (int k = 0; k < 288; ++k) s += gi[k] * W1[t * 288 + k];
  h1[t] = fmaxf(s, 0.f);
  __syncthreads();
  for (int j = t; j < 512; j += 256) {
    float s2 = b2[j];
    for (int k = 0; k < 256; ++k) s2 += h1[k] * W2[j * 256 + k];
    h2ws[j] = fmaxf(s2, 0.f);
  }
}

// ---------------- K2: big GEMV gen_states = W3 @ h2 + b3 -----------------
// One wave per output row; 512MB streamed with b128 loads.
__global__ void k2_gen_states(const float* __restrict__ W3, const float* __restrict__ b3,
                              const float* __restrict__ h2ws, float* __restrict__ gs) {
  __shared__ __align__(16) float sh2[512];
  int t = threadIdx.x;
  sh2[t] = h2ws[t];
  sh2[t + 256] = h2ws[t + 256];
  __syncthreads();
  int wave = t >> 5, lane = t & 31;
  int row = blockIdx.x * 8 + wave;
  const float4* wrow = (const float4*)(W3 + (size_t)row * 512);
  const float4* h4 = (const float4*)sh2;
  __builtin_prefetch(wrow + 128, 0, 0);  // global_prefetch_b8: next row
  float acc = 0.f;
#pragma unroll
  for (int j = 0; j < 4; ++j) {
    float4 w = wrow[lane + 32 * j];
    float4 h = h4[lane + 32 * j];
    acc += w.x * h.x + w.y * h.y + w.z * h.z + w.w * h.w;
  }
  acc = warp_sum(acc);
  if (lane == 0) gs[row] = acc + b3[row];
}

// ---------------- K3/K4: faction means + global opinion ------------------
__global__ void k3_fmean(const float* __restrict__ gs, float* __restrict__ fmean) {
  int f = blockIdx.x, d = threadIdx.x;
  float s = 0.f;
  for (int c = 0; c < 128; ++c) s += gs[(f * 128 + c) * 256 + d];
  fmean[f * 256 + d] = s * (1.f / 128.f);
}

__global__ void k4_gmean(const float* __restrict__ fmean, float* __restrict__ gmean) {
  int d = threadIdx.x;
  float s = 0.f;
#pragma unroll
  for (int f = 0; f < 8; ++f) s += fmean[f * 256 + d];
  gmean[d] = s * 0.125f;
}

// ---------------- K5: sync/debate + new_cell_hiddens + H -----------------
__global__ void k5_sync(const float* __restrict__ gs, const float* __restrict__ fmean,
                        const float* __restrict__ gmean, const float* __restrict__ ch,
                        const int* __restrict__ step, float* __restrict__ newcell,
                        float* __restrict__ Hws) {
  int cell = blockIdx.x, d = threadIdx.x;
  int f = cell >> 7, c = cell & 127;
  float v = 0.85f * gs[cell * 256 + d] + 0.15f * fmean[f * 256 + d];
  if (step[0] > 5 && c < 32) v = 0.85f * v + 0.15f * gmean[d];
  if (cell < 32) Hws[cell * 256 + d] = v;
  newcell[cell * 256 + d] = 0.7f * v + 0.3f * ch[cell * 256 + d];
}

// ---------------- K6: heads + GRU via WMMA (one 512-thread block) --------
// LDS carve (floats): outA[8192] | outG[8192] | gih[24576] (t1a/t1g overlay gih)
__global__ void k6_heads_gru(const float* __restrict__ x, const float* __restrict__ Hws,
                             const float* __restrict__ eaW1, const float* __restrict__ eab1,
                             const float* __restrict__ eaW2, const float* __restrict__ eab2,
                             const float* __restrict__ egW1, const float* __restrict__ egb1,
                             const float* __restrict__ egW2, const float* __restrict__ egb2,
                             const float* __restrict__ Wih, const float* __restrict__ bih,
                             const float* __restrict__ Whh, const float* __restrict__ bhh,
                             float* __restrict__ outs_ws, float* __restrict__ tens_ws,
                             float* __restrict__ d_out) {
  extern __shared__ float sm[];
  float* outA = sm;            // 32x256 (a, then outputs)
  float* outG = sm + 8192;     // 32x256 (g, then h_n)
  float* gih  = sm + 16384;    // 32x768 (sums for r,z; pure i_n in cols 512..767)
  float* t1a  = gih;           // 32x128 (dead before gih written)
  float* t1g  = gih + 4096;    // 32x128
  __shared__ float tens[32];

  const int tid = threadIdx.x;           // 512 threads = 16 waves
  const int wave = tid >> 5, lane = tid & 31;
  const int half = lane >> 4, l = lane & 15;

  // ---- layer 1: t1 = relu([x|H] @ {ea,eg}W1^T + b1)  (M=32,K=512,N=128)
  for (int t = wave; t < 32; t += 16) {
    int side = t >> 4, tt = t & 15;
    int mt = tt & 1, nt = tt >> 1;
    const float* W = side ? egW1 : eaW1;
    int arow = mt * 16 + l, bcol = nt * 16 + l;
    v8f c = {};
    for (int k = 0; k < 512; k += 4) {
      int ka = k + half * 2;
      v2f a, b;
      a.x = (ka < 256) ? x[ka] : Hws[arow * 256 + ka - 256];
      a.y = (ka + 1 < 256) ? x[ka + 1] : Hws[arow * 256 + ka + 1 - 256];
      b.x = W[bcol * 512 + ka];
      b.y = W[bcol * 512 + ka + 1];
      c = wmma4(a, b, c);
    }
    float bv = (side ? egb1 : eab1)[bcol];
    float* D = side ? t1g : t1a;
#pragma unroll
    for (int r = 0; r < 8; ++r)
      D[(mt * 16 + half * 8 + r) * 128 + bcol] = fmaxf(c[r] + bv, 0.f);
  }
  __syncthreads();

  // ---- layer 2: a/g = t1 @ {ea,eg}W2^T + b2  (M=32,K=128,N=256)
  for (int t = wave; t < 64; t += 16) {
    int side = t >> 5, tt = t & 31;
    int mt = tt & 1, nt = tt >> 1;
    const float* A = side ? t1g : t1a;
    const float* W = side ? egW2 : eaW2;
    int arow = mt * 16 + l, bcol = nt * 16 + l;
    v8f c = {};
    for (int k = 0; k < 128; k += 4) {
      int ka = k + half * 2;
      v2f a, b;
      a.x = A[arow * 128 + ka];     a.y = A[arow * 128 + ka + 1];
      b.x = W[bcol * 128 + ka];     b.y = W[bcol * 128 + ka + 1];
      c = wmma4(a, b, c);
    }
    float bv = (side ? egb2 : eab2)[bcol];
    float* D = side ? outG : outA;
#pragma unroll
    for (int r = 0; r < 8; ++r)
      D[(mt * 16 + half * 8 + r) * 256 + bcol] = c[r] + bv;
  }
  __syncthreads();

  // ---- outputs = a - g (keep in outA, mirror to ws)
  for (int i = tid; i < 8192; i += 512) {
    float v = outA[i] - outG[i];
    outA[i] = v;
    outs_ws[i] = v;
  }
  __syncthreads();

  // ---- tensions[i] = mean(outputs[i]^2)
  for (int r = wave; r < 32; r += 16) {
    float s = 0.f;
    for (int d = lane; d < 256; d += 32) { float v = outA[r * 256 + d]; s += v * v; }
    s = warp_sum(s) * (1.f / 256.f);
    if (lane == 0) { tens[r] = s; tens_ws[r] = s; }
  }
  __syncthreads();

  // ---- gih = [outputs|tension] @ Wih^T + bih  (M=32,K=256(+1),N=768)
  for (int t = wave; t < 96; t += 16) {
    int mt = t & 1, nt = t >> 1;
    int arow = mt * 16 + l, j = nt * 16 + l;
    v8f c = {};
    for (int k = 0; k < 256; k += 4) {
      int ka = k + half * 2;
      v2f a, b;
      a.x = outA[arow * 256 + ka];      a.y = outA[arow * 256 + ka + 1];
      b.x = Wih[j * 257 + ka];          b.y = Wih[j * 257 + ka + 1];
      c = wmma4(a, b, c);
    }
    float wlast = Wih[j * 257 + 256];
    float bv = bih[j];
#pragma unroll
    for (int r = 0; r < 8; ++r) {
      int i = mt * 16 + half * 8 + r;
      gih[i * 768 + j] = c[r] + tens[i] * wlast + bv;
    }
  }
  __syncthreads();

  // ---- ghh = H @ Whh^T + bhh; fold r/z halves into gih, keep h_n apart
  for (int t = wave; t < 96; t += 16) {
    int mt = t & 1, nt = t >> 1;
    int arow = mt * 16 + l, j = nt * 16 + l;
    v8f c = {};
    for (int k = 0; k < 256; k += 4) {
      int ka = k + half * 2;
      v2f a, b;
      a.x = Hws[arow * 256 + ka];       a.y = Hws[arow * 256 + ka + 1];
      b.x = Whh[j * 256 + ka];          b.y = Whh[j * 256 + ka + 1];
      c = wmma4(a, b, c);
    }
    float bv = bhh[j];
#pragma unroll
    for (int r = 0; r < 8; ++r) {
      int i = mt * 16 + half * 8 + r;
      float v = c[r] + bv;
      if (j < 512) gih[i * 768 + j] += v;        // (ir+hr), (iz+hz)
      else         outG[i * 256 + (j - 512)] = v; // h_n
    }
  }
  __syncthreads();

  // ---- GRU gates -> new_hiddens into d_out[259..]
  for (int idx = tid; idx < 8192; idx += 512) {
    int i = idx >> 8, d = idx & 255;
    float r = 1.f / (1.f + __expf(-gih[i * 768 + d]));
    float z = 1.f / (1.f + __expf(-gih[i * 768 + 256 + d]));
    float n = tanhf(gih[i * 768 + 512 + d] + r * outG[i * 256 + d]);
    d_out[259 + idx] = (1.f - z) * n + z * Hws[i * 256 + d];
  }
}

// ---------------- K7: fused dual discriminator layer-1 GEMV --------------
// disc_W1 (512 x 262144) read ONCE; both inputs dotted per row.
__global__ void k7_disc1(const float* __restrict__ W1, const float* __restrict__ b1,
                         const float* __restrict__ realv, const float* __restrict__ fakev,
                         float* __restrict__ d1r, float* __restrict__ d1f) {
  int row = blockIdx.x, t = threadIdx.x;
  const float4* w = (const float4*)(W1 + (size_t)row * FLATD);
  const float4* r4 = (const float4*)realv;
  const float4* f4 = (const float4*)fakev;
  float ar = 0.f, af = 0.f;
  for (int i = t; i < FLATD / 4; i += 256) {
    __builtin_prefetch(&w[i + 512], 0, 0);
    float4 wv = w[i];
    float4 rv = r4[i];
    float4 fv = f4[i];
    ar += wv.x * rv.x + wv.y * rv.y + wv.z * rv.z + wv.w * rv.w;
    af += wv.x * fv.x + wv.y * fv.y + wv.z * fv.z + wv.w * fv.w;
  }
  __shared__ float sr[8], sf[8];
  ar = warp_sum(ar);
  af = warp_sum(af);
  int wave = t >> 5, lane = t & 31;
  if (lane == 0) { sr[wave] = ar; sf[wave] = af; }
  __syncthreads();
  if (t == 0) {
    float a = b1[row], f = b1[row];
#pragma unroll
    for (int wv = 0; wv < 8; ++wv) { a += sr[wv]; f += sf[wv]; }
    d1r[row] = (a > 0.f) ? a : 0.2f * a;   // leaky_relu(0.2)
    d1f[row] = (f > 0.f) ? f : 0.2f * f;
  }
}

// ---------------- K8: discriminator tail (layers 2,3 + sigmoid) ----------
__global__ void k8_disc_tail(const float* __restrict__ d1r, const float* __restrict__ d1f,
                             const float* __restrict__ W2, const float* __restrict__ b2,
                             const float* __restrict__ W3, const float* __restrict__ b3,
                             float* __restrict__ d_out) {
  __shared__ float s1r[512], s1f[512];
  __shared__ float redr[8], redf[8];
  int t = threadIdx.x;  // 256
  s1r[t] = d1r[t]; s1r[t + 256] = d1r[t + 256];
  s1f[t] = d1f[t]; s1f[t + 256] = d1f[t + 256];
  __syncthreads();
  float ar = b2[t], af = b2[t];
  for (int k = 0; k < 512; ++k) {
    float w = W2[t * 512 + k];
    ar += w * s1r[k];
    af += w * s1f[k];
  }
  ar = (ar > 0.f) ? ar : 0.2f * ar;
  af = (af > 0.f) ? af : 0.2f * af;
  float pr = warp_sum(ar * W3[t]);
  float pf = warp_sum(af * W3[t]);
  int wave = t >> 5, lane = t & 31;
  if (lane == 0) { redr[wave] = pr; redf[wave] = pf; }
  __syncthreads();
  if (t == 0) {
    float sr = b3[0], sf = b3[0];
#pragma unroll
    for (int w = 0; w < 8; ++w) { sr += redr[w]; sf += redf[w]; }
    d_out[257] = 1.f / (1.f + __expf(-sr));  // d_real
    d_out[258] = 1.f / (1.f + __expf(-sf));  // d_fake
  }
}

// ---------------- K9: softmax over tensions, weighted combine ------------
__global__ void k9_combine(const float* __restrict__ tens, const float* __restrict__ outs,
                           float* __restrict__ d_out) {
  __shared__ float w[32];
  __shared__ float mt_s;
  int t = threadIdx.x;  // 256
  if (t == 0) {
    float mx = tens[0];
    for (int i = 1; i < 32; ++i) mx = fmaxf(mx, tens[i]);
    float e[32], s = 0.f, mean = 0.f;
    for (int i = 0; i < 32; ++i) {
      e[i] = __expf(tens[i] - mx);
      s += e[i];
      mean += tens[i];
    }
    mt_s = mean * (1.f / 32.f);
    float inv = 1.f / s;
    for (int i = 0; i < 32; ++i) w[i] = e[i] * inv;
  }
  __syncthreads();
  float acc = 0.f;
  for (int i = 0; i < 32; ++i) acc += w[i] * outs[i * 256 + t];
  d_out[t] = acc;
  if (t == 0) d_out[256] = mt_s;
}

// ---------------- launcher ------------------------------------------------
extern "C" void kernel_launch(void* const* d_in, const int* in_sizes, int n_in,
                              void* d_out, int out_size, void* d_ws, size_t ws_size,
                              hipStream_t stream) {
  const float* x      = (const float*)d_in[0];
  const float* noise  = (const float*)d_in[1];
  const float* ch     = (const float*)d_in[2];
  const float* genW1  = (const float*)d_in[3];
  const float* genb1  = (const float*)d_in[4];
  const float* genW2  = (const float*)d_in[5];
  const float* genb2  = (const float*)d_in[6];
  const float* genW3  = (const float*)d_in[7];
  const float* genb3  = (const float*)d_in[8];
  const float* discW1 = (const float*)d_in[9];
  const float* discb1 = (const float*)d_in[10];
  const float* discW2 = (const float*)d_in[11];
  const float* discb2 = (const float*)d_in[12];
  const float* discW3 = (const float*)d_in[13];
  const float* discb3 = (const float*)d_in[14];
  const float* eaW1   = (const float*)d_in[15];
  const float* eab1   = (const float*)d_in[16];
  const float* eaW2   = (const float*)d_in[17];
  const float* eab2   = (const float*)d_in[18];
  const float* egW1   = (const float*)d_in[19];
  const float* egb1   = (const float*)d_in[20];
  const float* egW2   = (const float*)d_in[21];
  const float* egb2   = (const float*)d_in[22];
  const float* Wih    = (const float*)d_in[23];
  const float* bih    = (const float*)d_in[24];
  const float* Whh    = (const float*)d_in[25];
  const float* bhh    = (const float*)d_in[26];
  const int*   step   = (const int*)d_in[27];

  float* ws = (float*)d_ws;
  float* h2      = ws;            // 512
  float* gs      = ws + 512;      // 262144 (gen_states)
  float* fmean   = ws + 262656;   // 2048
  float* gmean   = ws + 264704;   // 256
  float* newcell = ws + 264960;   // 262144
  float* Hws     = ws + 527104;   // 8192 (synced[:32])
  float* outs    = ws + 535296;   // 8192
  float* tns     = ws + 543488;   // 32
  float* d1r     = ws + 543520;   // 512
  float* d1f     = ws + 544032;   // 512

  float* out = (float*)d_out;

  k1_gen_front<<<1, 256, 0, stream>>>(x, noise, genW1, genb1, genW2, genb2, h2);
  k2_gen_states<<<FLATD / 8, 256, 0, stream>>>(genW3, genb3, h2, gs);
  k3_fmean<<<8, 256, 0, stream>>>(gs, fmean);
  k4_gmean<<<1, 256, 0, stream>>>(fmean, gmean);
  k5_sync<<<NCELLS, 256, 0, stream>>>(gs, fmean, gmean, ch, step, newcell, Hws);
  // dynamic LDS: (8192 + 8192 + 24576) floats = 160 KB (WGP has 320 KB)
  k6_heads_gru<<<1, 512, 40960 * sizeof(float), stream>>>(
      x, Hws, eaW1, eab1, eaW2, eab2, egW1, egb1, egW2, egb2,
      Wih, bih, Whh, bhh, outs, tns, out);
  k7_disc1<<<512, 256, 0, stream>>>(discW1, discb1, newcell, gs, d1r, d1f);
  k8_disc_tail<<<1, 256, 0, stream>>>(d1r, d1f, discW2, discb2, discW3, discb3, out);
  k9_combine<<<1, 256, 0, stream>>>(tns, outs, out);
}